// VectorQuantizer_16080357556666
// MI455X (gfx1250) — compile-verified
//
#include <hip/hip_runtime.h>
#include <hip/hip_bf16.h>

typedef float v2f __attribute__((ext_vector_type(2)));
typedef float v8f __attribute__((ext_vector_type(8)));

#define EMB_D   64
#define NUM_K   1024
#define HW      1024        // 32*32
#define N_TOTAL 65536       // 64*32*32
#define ROWS_PER_WAVE 32    // two 16-row WMMA tiles per wave

// --- kernel 1: zero the loss accumulator (must run every call: graph replay) ---
__global__ void vq_init_kernel(float* ws) {
    if (threadIdx.x == 0) ws[0] = 0.0f;
}

// --- kernel 2: ||e||^2 per code ---
__global__ void vq_enorm_kernel(const float* __restrict__ E, float* __restrict__ enorm) {
    int k = blockIdx.x * blockDim.x + threadIdx.x;
    if (k < NUM_K) {
        const float* row = E + (size_t)k * EMB_D;
        float s = 0.0f;
        #pragma unroll
        for (int d = 0; d < EMB_D; d += 4) {
            float4 v = *(const float4*)(row + d);
            s += v.x * v.x + v.y * v.y + v.z * v.z + v.w * v.w;
        }
        enorm[k] = s;
    }
}

// Per-16-row-tile epilogue: butterfly argmin across the 16 columns, gather the
// winning embedding rows, write output (coalesced along HW), return local loss.
__device__ __forceinline__ float vq_epilogue(
    float (&bv)[8], int (&bi)[8], int lane,
    const float* __restrict__ E, const float* __restrict__ zb,
    float* __restrict__ ob, int hw_base, int tbase)
{
    // C layout: lane = column (code), VGPR slot j = rows {j, j+8} per half.
    // Butterfly min within each 16-lane half; tie-break on lower code index.
    #pragma unroll
    for (int j = 0; j < 8; ++j) {
        #pragma unroll
        for (int mask = 8; mask >= 1; mask >>= 1) {
            float ov = __shfl_xor(bv[j], mask, 32);
            int   oi = __shfl_xor(bi[j], mask, 32);
            if (ov < bv[j] || (ov == bv[j] && oi < bv[j] - bv[j] + bi[j] ? (ov == bv[j] && oi < bi[j]) : false)) { }
            if (ov < bv[j] || (ov == bv[j] && oi < bi[j])) { bv[j] = ov; bi[j] = oi; }
        }
    }
    // lane l wants row (l & 15)'s code: slot (l & 7), pulled from the right half.
    int code_sel = bi[0];
    #pragma unroll
    for (int j = 1; j < 8; ++j)
        code_sel = ((lane & 7) == j) ? bi[j] : code_sel;
    const int src_lane = (lane & 8) ? (16 + (lane & 7)) : (lane & 7);
    const int row_code = __shfl(code_sel, src_lane, 32);

    const float* emy = E + (size_t)row_code * EMB_D;
    const int r    = lane & 15;
    const int half = lane >> 4;
    float lsum = 0.0f;
    #pragma unroll 4
    for (int it = 0; it < 32; ++it) {
        int d = it * 2 + half;
        float ev = emy[d];
        size_t off = (size_t)d * HW + (size_t)(hw_base + tbase + r);
        float zv = zb[off];
        float df = ev - zv;
        lsum += df * df;
        ob[off] = ev;          // lanes 0-15 / 16-31: two contiguous 64B runs
    }
    return lsum;
}

// --- kernel 3: main WMMA distance + argmin + gather/write + loss ---
__global__ __launch_bounds__(256) void vq_main_kernel(
    const float* __restrict__ Z,       // [64, 64, 32, 32] = [B, D, HW]
    const float* __restrict__ E,       // [1024, 64]
    const float* __restrict__ enorm,   // [1024]
    float* __restrict__ out,           // [B, D, HW]
    float* __restrict__ loss_acc)
{
    const int lane = threadIdx.x & 31;
    const int wave = blockIdx.x * (blockDim.x >> 5) + (threadIdx.x >> 5);
    const int row_base = wave * ROWS_PER_WAVE;     // 32 | 1024: never crosses b
    const int b       = row_base >> 10;
    const int hw_base = row_base & 1023;

    const int m    = lane & 15;                    // M row (A) / N col (B) within tile
    const int koff = (lane >> 4) << 1;             // low half: K{0,1}; high half: K{2,3}
    const int hw0  = hw_base + m;                  // tile 0 rows
    const int hw1  = hw0 + 16;                     // tile 1 rows

    const float* zb = Z + (size_t)b * (EMB_D * HW);
    float*       ob = out + (size_t)b * (EMB_D * HW);

    // ---- preload A fragments for BOTH 16-row tiles (16 k-steps each) ----
    // A layout: lanes 0-15 hold M=lane, VGPR0=K0,VGPR1=K1; lanes 16-31: K2,K3
    v2f a0[16], a1[16];
    #pragma unroll
    for (int t = 0; t < 16; ++t) {
        int d0 = 4 * t + koff;
        a0[t].x = zb[(size_t)(d0    ) * HW + hw0];
        a0[t].y = zb[(size_t)(d0 + 1) * HW + hw0];
        a1[t].x = zb[(size_t)(d0    ) * HW + hw1];
        a1[t].y = zb[(size_t)(d0 + 1) * HW + hw1];
    }

    float bv0[8], bv1[8];
    int   bi0[8], bi1[8];
    #pragma unroll
    for (int j = 0; j < 8; ++j) {
        bv0[j] = 3.4e38f; bi0[j] = 0;
        bv1[j] = 3.4e38f; bi1[j] = 0;
    }

    const int c = lane & 15;                        // column = code-within-tile
    for (int cb = 0; cb < NUM_K; cb += 16) {
        const float* erow = E + (size_t)(cb + c) * EMB_D + koff;
        if (cb + 16 < NUM_K)
            __builtin_prefetch(E + (size_t)(cb + 16 + c) * EMB_D, 0, 0);

        // B layout (4x16 KxN): lanes 0-15 = N, VGPR0=K0,VGPR1=K1; lanes 16-31: K2,K3
        v2f bf[16];
        #pragma unroll
        for (int t = 0; t < 16; ++t) {
            bf[t].x = erow[4 * t];
            bf[t].y = erow[4 * t + 1];
        }

        // Two independent WMMA accumulation chains share one B tile:
        // 16 B loads amortized over 32 WMMAs, and the chains interleave on XDL.
        v8f acc0 = {};
        v8f acc1 = {};
        #pragma unroll
        for (int t = 0; t < 16; ++t) {
            acc0 = __builtin_amdgcn_wmma_f32_16x16x4_f32(
                false, a0[t], false, bf[t], (short)0, acc0, false, false);
            acc1 = __builtin_amdgcn_wmma_f32_16x16x4_f32(
                false, a1[t], false, bf[t], (short)0, acc1, false, false);
        }

        const float en   = enorm[cb + c];           // ||e||^2, constant per column
        const int   code = cb + c;
        #pragma unroll
        for (int j = 0; j < 8; ++j) {
            float s0 = en - 2.0f * acc0[j];         // ||z||^2 constant per row: drop it
            float s1 = en - 2.0f * acc1[j];
            if (s0 < bv0[j]) { bv0[j] = s0; bi0[j] = code; }
            if (s1 < bv1[j]) { bv1[j] = s1; bi1[j] = code; }
        }
    }

    float lsum = 0.0f;
    lsum += vq_epilogue(bv0, bi0, lane, E, zb, ob, hw_base, 0);
    lsum += vq_epilogue(bv1, bi1, lane, E, zb, ob, hw_base, 16);

    // wave-reduce loss, one atomic per wave
    #pragma unroll
    for (int mask = 16; mask >= 1; mask >>= 1)
        lsum += __shfl_xor(lsum, mask, 32);
    if (lane == 0) atomicAdd(loss_acc, lsum);
}

// --- kernel 4: finalize loss = 1.25 * sum / (N * D) ---
__global__ void vq_finalize_kernel(const float* __restrict__ ws, float* __restrict__ out_loss) {
    if (threadIdx.x == 0)
        out_loss[0] = ws[0] * (1.25f / (float)(N_TOTAL * EMB_D));
}

extern "C" void kernel_launch(void* const* d_in, const int* in_sizes, int n_in,
                              void* d_out, int out_size, void* d_ws, size_t ws_size,
                              hipStream_t stream) {
    const float* Z = (const float*)d_in[0];   // [64, 64, 32, 32] f32
    const float* E = (const float*)d_in[1];   // [1024, 64] f32
    float* out = (float*)d_out;               // 64*64*32*32 quantized + 1 loss
    float* ws  = (float*)d_ws;                // ws[0] = loss acc; ws[16..16+1023] = ||e||^2

    vq_init_kernel<<<1, 32, 0, stream>>>(ws);
    vq_enorm_kernel<<<(NUM_K + 255) / 256, 256, 0, stream>>>(E, ws + 16);

    const int waves = N_TOTAL / ROWS_PER_WAVE;  // 2048 waves, 32 rows each
    const int wpb   = 256 / 32;                 // 8 waves per 256-thread block
    vq_main_kernel<<<waves / wpb, 256, 0, stream>>>(Z, E, ws + 16, out, ws);

    vq_finalize_kernel<<<1, 32, 0, stream>>>(ws, out + (out_size - 1));
}